// model_19052474925082
// MI455X (gfx1250) — compile-verified
//
#include <hip/hip_runtime.h>

#define THREADS 1024
#define NCAND   15376   // 5000 + 5000 + 4096 + 1024 + 256
#define NSORT   16384
#define NDETS   10000

// ---------------- CDNA5 async global->LDS path ----------------
#if defined(__HIP_DEVICE_COMPILE__)
#if __has_builtin(__builtin_amdgcn_global_load_async_to_lds_b32) && \
    __has_builtin(__builtin_amdgcn_s_wait_asynccnt)
#define HAVE_ASYNC_LDS 1
#endif
#if __has_builtin(__builtin_amdgcn_global_load_async_to_lds_b128)
#define HAVE_ASYNC_LDS_B128 1
#endif
#endif

typedef int v4i __attribute__((vector_size(16)));
typedef __attribute__((address_space(1))) int as1_int;   // global, scalar
typedef __attribute__((address_space(3))) int as3_int;   // LDS, scalar
typedef __attribute__((address_space(1))) v4i as1_v4i;   // global, 16B vector
typedef __attribute__((address_space(3))) v4i as3_v4i;   // LDS, 16B vector

__device__ __forceinline__ void g2l_async_f32(const float* g, float* l) {
#ifdef HAVE_ASYNC_LDS
  // (global src, lds dst, imm offset, imm cpol)
  __builtin_amdgcn_global_load_async_to_lds_b32((as1_int*)g, (as3_int*)l, 0, 0);
#else
  *l = *g;
#endif
}

// Copy n floats (n multiple of 4, both pointers 16B-aligned) global->LDS,
// block-strided, using the widest async-DMA op available.
__device__ __forceinline__ void stage_heat(const float* __restrict__ g,
                                           float* __restrict__ l, int n, int tid) {
#ifdef HAVE_ASYNC_LDS_B128
  const int nvec = n >> 2;
  for (int i = tid; i < nvec; i += THREADS)
    __builtin_amdgcn_global_load_async_to_lds_b128((as1_v4i*)(g + 4 * (size_t)i),
                                                   (as3_v4i*)(l + 4 * (size_t)i), 0, 0);
#else
  for (int i = tid; i < n; i += THREADS) g2l_async_f32(g + i, l + i);
#endif
}

__device__ __forceinline__ void async_fence_and_barrier() {
#ifdef HAVE_ASYNC_LDS
  __builtin_amdgcn_s_wait_asynccnt(0);
#endif
  __syncthreads();
}

// ---------------- row emission (shared by all levels) ----------------
__device__ __forceinline__ void emit_row(float* __restrict__ cand, int b, int level_off,
                                         unsigned slot, int p, float sc,
                                         const float* __restrict__ tlb,
                                         const float* __restrict__ brb,
                                         int n, int w, float wsf, float hsf, float levf) {
  const int y = p / w;
  const int x = p - y * w;
  const float fx = (float)x, fy = (float)y;
  // BASE_LAYER_RANGE/16 -> min=1.5, max=6.0 for both axes: span 4.5, center 3.75
  const float t0 = tlb[p], t1 = tlb[n + p];
  const float b0 = brb[p], b1 = brb[n + p];
  const float tlx = fx - (4.5f * t0 + 3.75f);
  const float tly = fy - (4.5f * t1 + 3.75f);
  const float brx = fx + (4.5f * b0 + 3.75f);
  const float bry = fy + (4.5f * b1 + 3.75f);
  if (brx < tlx || bry < tly) sc = -1.0f;
  float* row = cand + ((size_t)b * NCAND + (size_t)level_off + slot) * 7;
  row[0] = sc;
  row[1] = tlx * wsf;
  row[2] = tly * hsf;
  row[3] = brx * wsf;
  row[4] = bry * hsf;
  row[5] = 0.0f;
  row[6] = levf;
}

// ---------------- levels 0/1: exact top-K via 3-pass LDS radix select ----------------
__global__ __launch_bounds__(THREADS, 1)
void select_topk_kernel(const float* __restrict__ heat, const float* __restrict__ tl,
                        const float* __restrict__ br, float* __restrict__ cand,
                        int h, int w, int K, int level_off, int level) {
  const int n = h * w;
  const int b = blockIdx.x;
  const int tid = threadIdx.x;

  extern __shared__ unsigned char smem_raw[];
  float*    s_heat  = (float*)smem_raw;                         // n floats
  unsigned* s_hist  = (unsigned*)(smem_raw + (size_t)n * 4);    // 2048
  unsigned* s_gcnt  = s_hist + 2048;                            // 1024
  unsigned* s_ecnt  = s_gcnt + THREADS;                         // 1024
  unsigned* s_gscan = s_ecnt + THREADS;                         // 1024
  unsigned* s_escan = s_gscan + THREADS;                        // 1024
  unsigned* s_misc  = s_escan + THREADS;                        // 8

  // Stage the entire level heatmap into LDS via async DMA (ASYNCcnt tracked).
  stage_heat(heat + (size_t)b * n, s_heat, n, tid);
  if (tid == 0) s_misc[1] = (unsigned)K;
  async_fence_and_barrier();

  // Radix select the K-th largest bit pattern. heat in [0,1): uint bits are
  // order-isomorphic to float order.
  unsigned prefix = 0, pmask = 0;
  const int shifts[3] = {21, 10, 0};
  const int nb_arr[3] = {2048, 2048, 1024};
#pragma unroll 1
  for (int pass = 0; pass < 3; ++pass) {
    const int shift = shifts[pass];
    const unsigned bm = (unsigned)nb_arr[pass] - 1u;
    for (int i = tid; i < 2048; i += THREADS) s_hist[i] = 0u;
    __syncthreads();
    for (int i = tid; i < n; i += THREADS) {
      const unsigned u = __float_as_uint(s_heat[i]);
      if ((u & pmask) == prefix) atomicAdd(&s_hist[(u >> shift) & bm], 1u);
    }
    __syncthreads();
    if (tid == 0) {
      const unsigned remK = s_misc[1];
      unsigned cum = 0, sel = 0;
      for (int bin = (int)bm; bin >= 0; --bin) {
        const unsigned c = s_hist[bin];
        if (cum + c >= remK) { sel = (unsigned)bin; s_misc[1] = remK - cum; break; }
        cum += c;
      }
      s_misc[0] = sel;
    }
    __syncthreads();
    prefix |= s_misc[0] << shift;
    pmask  |= bm << (unsigned)shift;
    __syncthreads();
  }
  // prefix == exact bit pattern of the K-th largest value,
  // s_misc[1] == number of elements equal to it that must be kept.

  // Stable compaction: per-thread (strided-subset) counts + scan.
  unsigned g = 0, e = 0;
  for (int i = tid; i < n; i += THREADS) {
    const unsigned u = __float_as_uint(s_heat[i]);
    g += (u > prefix);
    e += (u == prefix);
  }
  s_gcnt[tid] = g;
  s_ecnt[tid] = e;
  __syncthreads();
  if (tid == 0) {
    unsigned ga = 0, ea = 0;
    for (int t = 0; t < THREADS; ++t) {
      const unsigned tg = s_gcnt[t]; s_gscan[t] = ga; ga += tg;
      const unsigned te = s_ecnt[t]; s_escan[t] = ea; ea += te;
    }
    s_misc[2] = ga;  // G = #strictly-greater
  }
  __syncthreads();

  const unsigned G = s_misc[2];
  const unsigned needE = s_misc[1];  // equals to keep; G + needE == K
  unsigned gofs = s_gscan[tid];
  unsigned eofs = s_escan[tid];
  const float wsf = 256.0f / (float)w;
  const float hsf = 256.0f / (float)h;
  const float levf = (float)level;
  const float* tlb = tl + (size_t)b * 2 * n;
  const float* brb = br + (size_t)b * 2 * n;

  for (int i = tid; i < n; i += THREADS) {
    const float v = s_heat[i];
    const unsigned u = __float_as_uint(v);
    if (u > prefix) {
      emit_row(cand, b, level_off, gofs, i, v, tlb, brb, n, w, wsf, hsf, levf);
      ++gofs;
    } else if (u == prefix) {
      if (eofs < needE)
        emit_row(cand, b, level_off, G + eofs, i, v, tlb, brb, n, w, wsf, hsf, levf);
      ++eofs;
    }
  }
}

// ---------------- levels 2-4: K == h*w, every position is a candidate ----------------
__global__ void direct_rows_kernel(const float* __restrict__ heat, const float* __restrict__ tl,
                                   const float* __restrict__ br, float* __restrict__ cand,
                                   int h, int w, int level_off, int level) {
  const int n = h * w;
  const int b = blockIdx.y;
  const int p = blockIdx.x * blockDim.x + threadIdx.x;
  if (p >= n) return;
  const float sc = heat[(size_t)b * n + p];
  emit_row(cand, b, level_off, (unsigned)p, p, sc,
           tl + (size_t)b * 2 * n, br + (size_t)b * 2 * n,
           n, w, 256.0f / (float)w, 256.0f / (float)h, (float)level);
}

// ---------------- final: per-batch bitonic sort of 16384 keys in 128KB LDS ----------------
__global__ __launch_bounds__(THREADS, 1)
void sort_out_kernel(const float* __restrict__ cand, float* __restrict__ out) {
  const int b = blockIdx.x;
  const int tid = threadIdx.x;
  extern __shared__ unsigned char smem_raw2[];
  unsigned long long* s_keys = (unsigned long long*)smem_raw2;  // NSORT u64

  for (int j = tid; j < NSORT; j += THREADS) {
    unsigned long long key = 0ull;
    if (j < NCAND) {
      const float sc = cand[((size_t)b * NCAND + j) * 7];
      const unsigned u = __float_as_uint(sc);
      const unsigned m = (u & 0x80000000u) ? ~u : (u | 0x80000000u);  // order-preserving
      key = ((unsigned long long)m << 32) | (unsigned)(0xFFFFFFFFu - (unsigned)j);
    }
    s_keys[j] = key;  // padding (0) sinks below every real candidate (m >= 0x407FFFFF)
  }
  __syncthreads();

  for (unsigned k = 2; k <= NSORT; k <<= 1) {
    for (unsigned stride = k >> 1; stride > 0; stride >>= 1) {
      for (unsigned i = tid; i < NSORT; i += THREADS) {
        const unsigned ixj = i ^ stride;
        if (ixj > i) {
          const unsigned long long a = s_keys[i];
          const unsigned long long c = s_keys[ixj];
          const bool up = ((i & k) == 0);  // ascending segment
          if (up ? (a > c) : (a < c)) { s_keys[i] = c; s_keys[ixj] = a; }
        }
      }
      __syncthreads();
    }
  }

  // Top NDETS, descending score, ties -> lowest candidate index first.
  for (int p = tid; p < NDETS; p += THREADS) {
    const unsigned long long key = s_keys[NSORT - 1 - p];
    const unsigned j = 0xFFFFFFFFu - (unsigned)(key & 0xFFFFFFFFull);
    const float* src = cand + ((size_t)b * NCAND + j) * 7;
    float* dst = out + ((size_t)b * NDETS + p) * 7;
#pragma unroll
    for (int c = 0; c < 7; ++c) dst[c] = src[c];
  }
}

// ---------------- host launcher ----------------
extern "C" void kernel_launch(void* const* d_in, const int* in_sizes, int n_in,
                              void* d_out, int out_size, void* d_ws, size_t ws_size,
                              hipStream_t stream) {
  (void)in_sizes; (void)n_in; (void)out_size; (void)ws_size;
  // setup_inputs dict order: heat0, tl0, br0, heat1, tl1, br1, ...
  const float* heat[5];
  const float* tlp[5];
  const float* brp[5];
  for (int i = 0; i < 5; ++i) {
    heat[i] = (const float*)d_in[3 * i + 0];
    tlp[i]  = (const float*)d_in[3 * i + 1];
    brp[i]  = (const float*)d_in[3 * i + 2];
  }
  float* cand = (float*)d_ws;  // 64 * 15376 * 7 floats (~27.6 MB)

  const size_t extraLds = (size_t)(2048 + 4 * THREADS + 8) * 4;
  const size_t lds0 = (size_t)65536 * 4 + extraLds;  // ~280 KB (fits 320 KB WGP LDS)
  const size_t lds1 = (size_t)16384 * 4 + extraLds;
  (void)hipFuncSetAttribute((const void*)select_topk_kernel,
                            hipFuncAttributeMaxDynamicSharedMemorySize, (int)lds0);
  (void)hipFuncSetAttribute((const void*)sort_out_kernel,
                            hipFuncAttributeMaxDynamicSharedMemorySize, (int)(NSORT * 8));

  select_topk_kernel<<<64, THREADS, lds0, stream>>>(heat[0], tlp[0], brp[0], cand,
                                                    256, 256, 5000, 0, 0);
  select_topk_kernel<<<64, THREADS, lds1, stream>>>(heat[1], tlp[1], brp[1], cand,
                                                    128, 128, 5000, 5000, 1);
  direct_rows_kernel<<<dim3(16, 64), 256, 0, stream>>>(heat[2], tlp[2], brp[2], cand,
                                                       64, 64, 10000, 2);
  direct_rows_kernel<<<dim3(4, 64), 256, 0, stream>>>(heat[3], tlp[3], brp[3], cand,
                                                      32, 32, 14096, 3);
  direct_rows_kernel<<<dim3(1, 64), 256, 0, stream>>>(heat[4], tlp[4], brp[4], cand,
                                                      16, 16, 15120, 4);
  sort_out_kernel<<<64, THREADS, NSORT * 8, stream>>>(cand, (float*)d_out);
}